// SeedGCN_360777253129
// MI455X (gfx1250) — compile-verified
//
#include <hip/hip_runtime.h>
#include <hip/hip_bf16.h>

typedef float v2f __attribute__((ext_vector_type(2)));
typedef float v8f __attribute__((ext_vector_type(8)));

#define NN 100000
#define EE 1600000
#define DH 128

// ---------------------------------------------------------------- utilities
__global__ __launch_bounds__(256) void zero_kernel(float* p, long n) {
  long i = (long)blockIdx.x * 256 + threadIdx.x;
  if (i < n) p[i] = 0.f;
}

// deg[t][dst] += w  (w already carries the edge-type mask semantics: we route by type)
__global__ __launch_bounds__(256) void degree_kernel(const int* __restrict__ dst,
                                                     const float* __restrict__ ew,
                                                     const int* __restrict__ et,
                                                     float* __restrict__ deg0,
                                                     float* __restrict__ deg1) {
  int e = blockIdx.x * 256 + threadIdx.x;
  if (e >= EE) return;
  float w = ew[e];
  float* deg = (et[e] == 0) ? deg0 : deg1;
  unsafeAtomicAdd(deg + dst[e], w);
}

// dis = rsqrt(deg+1), inv = 1/(deg+1) for both edge types (dis overwrites deg)
__global__ __launch_bounds__(256) void norm_kernel(float* d0, float* i0, float* d1, float* i1) {
  int i = blockIdx.x * 256 + threadIdx.x;
  if (i >= NN) return;
  float v0 = d0[i] + 1.f;
  d0[i] = rsqrtf(v0);
  i0[i] = 1.f / v0;
  float v1 = d1[i] + 1.f;
  d1[i] = rsqrtf(v1);
  i1[i] = 1.f / v1;
}

// ------------------------------------------------- fp32 WMMA GEMM, K=128, 128 cols
// grid.x = N/16 ; block = 256 (8 waves) ; wave w computes the 16x16 tile at cols 16w.
__global__ __launch_bounds__(256) void gemm128_wmma(const float* __restrict__ A,
                                                    const float* __restrict__ W,
                                                    const float* __restrict__ bias,
                                                    float* __restrict__ out, int relu) {
  __shared__ float lds[16 * 132];  // padded pitch: conflict-free column reads
  const int m0 = blockIdx.x * 16;
  for (int idx = threadIdx.x; idx < 16 * 32; idx += 256) {
    int rrow = idx >> 5, cc = idx & 31;
    ((float4*)(lds + rrow * 132))[cc] = ((const float4*)(A + (size_t)(m0 + rrow) * DH))[cc];
  }
  __syncthreads();
  const int wave = threadIdx.x >> 5;
  const int lane = threadIdx.x & 31;
  const int r = lane & 15;
  const int hi = lane >> 4;     // 0: K pair {0,1}, 1: K pair {2,3}
  const int koff = hi * 2;
  const int col = wave * 16 + r;
  v8f acc = {};
#pragma unroll
  for (int k0 = 0; k0 < 128; k0 += 4) {
    v2f a, b;
    a.x = lds[r * 132 + k0 + koff];
    a.y = lds[r * 132 + k0 + koff + 1];
    b.x = W[(k0 + koff) * DH + col];
    b.y = W[(k0 + koff + 1) * DH + col];
    acc = __builtin_amdgcn_wmma_f32_16x16x4_f32(false, a, false, b, (short)0, acc, false, false);
  }
  float bv = bias ? bias[col] : 0.f;
#pragma unroll
  for (int rr = 0; rr < 8; ++rr) {
    int row = m0 + rr + hi * 8;
    float v = acc[rr] + bv;
    if (relu) v = fmaxf(v, 0.f);
    out[(size_t)row * DH + col] = v;
  }
}

// ------------------------------------------------- edge scatter: one wave per edge
__global__ __launch_bounds__(256) void scatter_kernel(const float* __restrict__ feat,
                                                      const int* __restrict__ src,
                                                      const int* __restrict__ dst,
                                                      const float* __restrict__ ew,
                                                      const int* __restrict__ et,
                                                      int ttype,
                                                      const float* __restrict__ dis,
                                                      float* __restrict__ agg) {
  int wid = (blockIdx.x * 256 + threadIdx.x) >> 5;
  int lane = threadIdx.x & 31;
  if (wid >= EE) return;
  if (et[wid] != ttype) return;
  float w = ew[wid];
  if (w == 0.f) return;
  int s = src[wid], d = dst[wid];
  float coef = dis[s] * w * dis[d];
  const float* fs = feat + (size_t)s * DH;
  float* ad = agg + (size_t)d * DH;
#pragma unroll
  for (int c = 0; c < 4; ++c) {
    unsafeAtomicAdd(ad + lane + 32 * c, coef * fs[lane + 32 * c]);
  }
}

// o = BN(agg + lin*inv + b) [+ relu] [+ residual]
__global__ __launch_bounds__(256) void post_kernel(const float* __restrict__ agg,
                                                   const float* __restrict__ lin,
                                                   const float* __restrict__ inv,
                                                   const float* __restrict__ b,
                                                   const float* __restrict__ gamma,
                                                   const float* __restrict__ beta,
                                                   const float* __restrict__ rm,
                                                   const float* __restrict__ rv,
                                                   const float* __restrict__ resid,
                                                   float* __restrict__ out, int relu) {
  size_t i = (size_t)blockIdx.x * 256 + threadIdx.x;
  int colc = (int)(i & (DH - 1));
  int row = (int)(i >> 7);
  float o = agg[i] + lin[i] * inv[row] + b[colc];
  o = (o - rm[colc]) * gamma[colc] * rsqrtf(rv[colc] + 1e-5f) + beta[colc];
  if (relu) o = fmaxf(o, 0.f);
  if (resid) o += resid[i];
  out[i] = o;
}

// cvec[j] = pb1[j] + x0[seed]·pw1[256:384, j] + x1[seed]·pw1[384:512, j]
__global__ void cvec_kernel(const float* __restrict__ x0, const float* __restrict__ x1,
                            const int* __restrict__ seedp, const float* __restrict__ pw1,
                            const float* __restrict__ pb1, float* __restrict__ cvec) {
  int j = threadIdx.x;  // 64 threads
  int seed = seedp[0];
  const float* s0 = x0 + (size_t)seed * DH;
  const float* s1 = x1 + (size_t)seed * DH;
  float s = pb1[j];
  for (int k = 0; k < DH; ++k) {
    s += s0[k] * pw1[(256 + k) * 64 + j];
    s += s1[k] * pw1[(384 + k) * 64 + j];
  }
  cvec[j] = s;
}

// z1 = relu(x0@pw1[0:128] + x1@pw1[128:256] + cvec)   [N,64], K=256 via two halves
__global__ __launch_bounds__(256) void pred1_wmma(const float* __restrict__ x0,
                                                  const float* __restrict__ x1,
                                                  const float* __restrict__ pw1,
                                                  const float* __restrict__ cvec,
                                                  float* __restrict__ z1) {
  __shared__ float lds[32 * 132];
  const int m0 = blockIdx.x * 32;
  const int wave = threadIdx.x >> 5;
  const int lane = threadIdx.x & 31;
  const int r = lane & 15, hi = lane >> 4, koff = hi * 2;
  const int mt = wave >> 2, nt = wave & 3;  // 2 row-tiles x 4 col-tiles
  const int col = nt * 16 + r;
  v8f acc = {};
  const float* srcs[2] = {x0, x1};
  for (int half = 0; half < 2; ++half) {
    __syncthreads();
    for (int idx = threadIdx.x; idx < 32 * 32; idx += 256) {
      int rrow = idx >> 5, cc = idx & 31;
      ((float4*)(lds + rrow * 132))[cc] =
          ((const float4*)(srcs[half] + (size_t)(m0 + rrow) * DH))[cc];
    }
    __syncthreads();
    const float* Wp = pw1 + half * 128 * 64;
    for (int k0 = 0; k0 < 128; k0 += 4) {
      v2f a, b;
      a.x = lds[(mt * 16 + r) * 132 + k0 + koff];
      a.y = lds[(mt * 16 + r) * 132 + k0 + koff + 1];
      b.x = Wp[(k0 + koff) * 64 + col];
      b.y = Wp[(k0 + koff + 1) * 64 + col];
      acc = __builtin_amdgcn_wmma_f32_16x16x4_f32(false, a, false, b, (short)0, acc, false, false);
    }
  }
  float cv = cvec[col];
#pragma unroll
  for (int rr = 0; rr < 8; ++rr) {
    int row = m0 + mt * 16 + rr + hi * 8;
    z1[(size_t)row * 64 + col] = fmaxf(acc[rr] + cv, 0.f);
  }
}

// z2 = relu(z1@pw2 + b2); out = z2·pw3 + b3
__global__ __launch_bounds__(256) void pred23_wmma(const float* __restrict__ z1,
                                                   const float* __restrict__ pw2,
                                                   const float* __restrict__ pb2,
                                                   const float* __restrict__ pw3,
                                                   const float* __restrict__ pb3,
                                                   float* __restrict__ out) {
  __shared__ float ldsZ[32 * 68];
  __shared__ float ldsZ2[32 * 68];
  const int m0 = blockIdx.x * 32;
  for (int idx = threadIdx.x; idx < 32 * 16; idx += 256) {
    int rrow = idx >> 4, cc = idx & 15;
    ((float4*)(ldsZ + rrow * 68))[cc] = ((const float4*)(z1 + (size_t)(m0 + rrow) * 64))[cc];
  }
  __syncthreads();
  const int wave = threadIdx.x >> 5;
  const int lane = threadIdx.x & 31;
  const int r = lane & 15, hi = lane >> 4, koff = hi * 2;
  const int mt = wave >> 2, nt = wave & 3;
  const int col = nt * 16 + r;
  v8f acc = {};
#pragma unroll
  for (int k0 = 0; k0 < 64; k0 += 4) {
    v2f a, b;
    a.x = ldsZ[(mt * 16 + r) * 68 + k0 + koff];
    a.y = ldsZ[(mt * 16 + r) * 68 + k0 + koff + 1];
    b.x = pw2[(k0 + koff) * 64 + col];
    b.y = pw2[(k0 + koff + 1) * 64 + col];
    acc = __builtin_amdgcn_wmma_f32_16x16x4_f32(false, a, false, b, (short)0, acc, false, false);
  }
  float bv = pb2[col];
#pragma unroll
  for (int rr = 0; rr < 8; ++rr) {
    int lrow = mt * 16 + rr + hi * 8;
    ldsZ2[lrow * 68 + col] = fmaxf(acc[rr] + bv, 0.f);
  }
  __syncthreads();
  int row = threadIdx.x >> 3;  // 32 rows, 8 threads per row
  int seg = threadIdx.x & 7;
  float p = 0.f;
#pragma unroll
  for (int j = 0; j < 8; ++j) p += ldsZ2[row * 68 + seg * 8 + j] * pw3[seg * 8 + j];
  p += __shfl_down(p, 4);
  p += __shfl_down(p, 2);
  p += __shfl_down(p, 1);
  if (seg == 0) out[m0 + row] = p + pb3[0];
}

// ---------------------------------------------------------------- launcher
extern "C" void kernel_launch(void* const* d_in, const int* in_sizes, int n_in,
                              void* d_out, int out_size, void* d_ws, size_t ws_size,
                              hipStream_t stream) {
  const float* x = (const float*)d_in[0];
  const int* eidx = (const int*)d_in[1];
  const int* etype = (const int*)d_in[2];
  const float* ew = (const float*)d_in[3];
  const int* seedp = (const int*)d_in[4];
  const float* ft_w = (const float*)d_in[5];
  const float* ft_b = (const float*)d_in[6];
  const float* g_w[2] = {(const float*)d_in[7], (const float*)d_in[13]};
  const float* g_b[2] = {(const float*)d_in[8], (const float*)d_in[14]};
  const float* g_gamma[2] = {(const float*)d_in[9], (const float*)d_in[15]};
  const float* g_beta[2] = {(const float*)d_in[10], (const float*)d_in[16]};
  const float* g_rm[2] = {(const float*)d_in[11], (const float*)d_in[17]};
  const float* g_rv[2] = {(const float*)d_in[12], (const float*)d_in[18]};
  const float* pw1 = (const float*)d_in[19];
  const float* pb1 = (const float*)d_in[20];
  const float* pw2 = (const float*)d_in[21];
  const float* pb2 = (const float*)d_in[22];
  const float* pw3 = (const float*)d_in[23];
  const float* pb3 = (const float*)d_in[24];

  const int* srcI = eidx;
  const int* dstI = eidx + EE;

  float* ws = (float*)d_ws;
  float* h = ws;    ws += (size_t)NN * DH;
  float* lin = ws;  ws += (size_t)NN * DH;
  float* agg = ws;  ws += (size_t)NN * DH;
  float* hblk = ws; ws += (size_t)NN * DH;
  float* xb[2];
  xb[0] = ws;       ws += (size_t)NN * DH;
  xb[1] = ws;       ws += (size_t)NN * DH;
  float* d0 = ws;   ws += NN;
  float* i0 = ws;   ws += NN;
  float* d1 = ws;   ws += NN;
  float* i1 = ws;   ws += NN;
  float* cvec = ws; ws += 64;
  float* z1 = lin;  // lin is free by predictor time

  const int GEMM_BLOCKS = NN / 16;              // 6250
  const int POST_BLOCKS = (NN * DH) / 256;      // 50000
  const int SCAT_BLOCKS = EE / 8;               // 200000 (1 wave / edge)
  const int DEG_BLOCKS = EE / 256;              // 6250
  const int NORM_BLOCKS = (NN + 255) / 256;

  // degrees + norms (d0..i1 are contiguous: zero all 4N)
  zero_kernel<<<(4 * NN + 255) / 256, 256, 0, stream>>>(d0, (long)4 * NN);
  degree_kernel<<<DEG_BLOCKS, 256, 0, stream>>>(dstI, ew, etype, d0, d1);
  norm_kernel<<<NORM_BLOCKS, 256, 0, stream>>>(d0, i0, d1, i1);

  // h = relu(x @ ft_w + ft_b)
  gemm128_wmma<<<GEMM_BLOCKS, 256, 0, stream>>>(x, ft_w, ft_b, h, 1);

  for (int b = 0; b < 2; ++b) {
    const float* dis = (b == 0) ? d0 : d1;
    const float* inv = (b == 0) ? i0 : i1;
    // layer 0: no residual, relu
    gemm128_wmma<<<GEMM_BLOCKS, 256, 0, stream>>>(h, g_w[b], nullptr, lin, 0);
    zero_kernel<<<POST_BLOCKS, 256, 0, stream>>>(agg, (long)NN * DH);
    scatter_kernel<<<SCAT_BLOCKS, 256, 0, stream>>>(lin, srcI, dstI, ew, etype, b, dis, agg);
    post_kernel<<<POST_BLOCKS, 256, 0, stream>>>(agg, lin, inv, g_b[b], g_gamma[b], g_beta[b],
                                                 g_rm[b], g_rv[b], nullptr, hblk, 1);
    // layer 1: residual, no relu
    gemm128_wmma<<<GEMM_BLOCKS, 256, 0, stream>>>(hblk, g_w[b] + DH * DH, nullptr, lin, 0);
    zero_kernel<<<POST_BLOCKS, 256, 0, stream>>>(agg, (long)NN * DH);
    scatter_kernel<<<SCAT_BLOCKS, 256, 0, stream>>>(lin, srcI, dstI, ew, etype, b, dis, agg);
    post_kernel<<<POST_BLOCKS, 256, 0, stream>>>(agg, lin, inv, g_b[b] + DH, g_gamma[b] + DH,
                                                 g_beta[b] + DH, g_rm[b] + DH, g_rv[b] + DH,
                                                 hblk, xb[b], 0);
  }

  // predictor
  cvec_kernel<<<1, 64, 0, stream>>>(xb[0], xb[1], seedp, pw1, pb1, cvec);
  pred1_wmma<<<NN / 32, 256, 0, stream>>>(xb[0], xb[1], pw1, cvec, z1);
  pred23_wmma<<<NN / 32, 256, 0, stream>>>(z1, pw2, pb2, pw3, pb3, (float*)d_out);
}